// RGINConv_8280696947363
// MI455X (gfx1250) — compile-verified
//
#include <hip/hip_runtime.h>

#define N_NODES 50000
#define N_EDGES 156000
#define D 512
#define NREL 3

typedef __attribute__((ext_vector_type(2))) float v2f;
typedef __attribute__((ext_vector_type(8))) float v8f;

#define KC   32    // K-chunk staged in LDS
#define LDA  34    // A tile row stride (dwords): conflict-free for wave32 b64 reads
#define PSTR 160   // B pair-row stride (dwords): 160 % 64 == 32 -> lanes 16-31 hit opposite bank half

// ---------------------------------------------------------------------------
// agg = x  (element copy, float4 vectorized)
// ---------------------------------------------------------------------------
__global__ __launch_bounds__(256) void copy_kernel(const float* __restrict__ src,
                                                   float* __restrict__ dst, int n4) {
    int i = blockIdx.x * 256 + threadIdx.x;
    if (i < n4) ((float4*)dst)[i] = ((const float4*)src)[i];
}

// ---------------------------------------------------------------------------
// agg[dst] += x[src] for edges of relation `rel` (one block per edge)
// ---------------------------------------------------------------------------
__global__ __launch_bounds__(256) void scatter_kernel(const float* __restrict__ x,
                                                      const int* __restrict__ src_idx,
                                                      const int* __restrict__ dst_idx,
                                                      const int* __restrict__ etype,
                                                      int rel,
                                                      float* __restrict__ agg) {
    int e = blockIdx.x;
    if (etype[e] != rel) return;
    const float* xs = x   + (size_t)src_idx[e] * D;
    float*       ad = agg + (size_t)dst_idx[e] * D;
    for (int t = threadIdx.x; t < D; t += 256)
        atomicAdd(&ad[t], xs[t]);
}

// ---------------------------------------------------------------------------
// C[M,D] (+)= act(A[M,D] @ W[D,D] + bias)   via V_WMMA_F32_16X16X4_F32
// Block: 256 threads = 8 waves, computes a 128x64 tile.
// Wave w owns rows m0 + w*16 .. +15, all 64 cols (4 accumulators).
// B tile stored K-pair-interleaved in LDS so each B fragment is one b64 load.
// ---------------------------------------------------------------------------
template <bool RELU, bool ACC>
__global__ __launch_bounds__(256) void gemm_kernel(const float* __restrict__ A,
                                                   const float* __restrict__ W,
                                                   const float* __restrict__ bias,
                                                   float* __restrict__ C, int M) {
    __shared__ float sA[128 * LDA];          // 128 x KC, row-major, stride LDA
    __shared__ float sB[(KC / 2) * PSTR];    // pair p = k/2: dword addr p*PSTR + n*2 + (k&1)

    const int tid  = threadIdx.x;
    const int wave = tid >> 5;
    const int lane = tid & 31;
    const int half = lane >> 4;   // 0: lanes 0-15, 1: lanes 16-31
    const int l16  = lane & 15;

    const int m0   = blockIdx.y * 128;
    const int n0   = blockIdx.x * 64;
    const int wrow = wave * 16;   // wave's row strip inside the tile

    v8f acc[4] = {{}, {}, {}, {}};

    // B staging assignment: one (pair-row, 4-col group) per thread
    const int bp  = tid >> 4;          // 0..15 : K-pair row
    const int bc4 = (tid & 15) << 2;   // 0..60 : column group

    for (int k0 = 0; k0 < D; k0 += KC) {
        __syncthreads();   // previous chunk's compute done before restaging

        // ---- Stage A tile: 128 x KC (4096 floats, 4 x float4 per thread) ----
#pragma unroll
        for (int i = 0; i < 4; ++i) {
            int idx = tid + i * 256;
            int r   = idx >> 3;           // 0..127
            int c4  = (idx & 7) << 2;     // 0,4,..,28
            int gr  = m0 + r;
            gr      = gr < M ? gr : M - 1;   // clamp OOB rows (stores are masked)
            const float* gp = A + (size_t)gr * D + k0 + c4;
            float4 v = *(const float4*)gp;
            float* p = &sA[r * LDA + c4];
            p[0] = v.x; p[1] = v.y; p[2] = v.z; p[3] = v.w;
            if (k0 + KC < D) __builtin_prefetch(gp + KC, 0, 1);
        }
        // ---- Stage B tile: rows 2*bp, 2*bp+1, cols bc4..bc4+3, pair-interleaved ----
        {
            const float* g0 = W + (size_t)(k0 + 2 * bp) * D + n0 + bc4;
            float4 r0 = *(const float4*)g0;
            float4 r1 = *(const float4*)(g0 + D);
            float* p  = &sB[bp * PSTR + bc4 * 2];
            ((v2f*)p)[0] = (v2f){r0.x, r1.x};
            ((v2f*)p)[1] = (v2f){r0.y, r1.y};
            ((v2f*)p)[2] = (v2f){r0.z, r1.z};
            ((v2f*)p)[3] = (v2f){r0.w, r1.w};
            if (k0 + KC < D) __builtin_prefetch(g0 + (size_t)KC * D, 0, 1);
        }
        __syncthreads();

        // ---- Compute: 8 k-steps of 4; 1 A-b64 + 4 B-b64 feed 4 WMMAs ----
#pragma unroll
        for (int kk = 0; kk < KC; kk += 4) {
            // A frag (16x4): lanes 0-15 -> K=kk+0,kk+1 ; lanes 16-31 -> K=kk+2,kk+3
            v2f a = *(const v2f*)&sA[(wrow + l16) * LDA + kk + 2 * half];
            // B frag pair index for K = kk + 2*half (even)
            int p = (kk >> 1) + half;
#pragma unroll
            for (int c = 0; c < 4; ++c) {
                int bc = c * 16 + l16;
                v2f b  = *(const v2f*)&sB[p * PSTR + bc * 2];
                acc[c] = __builtin_amdgcn_wmma_f32_16x16x4_f32(false, a, false, b,
                                                               (short)0, acc[c], false, false);
            }
        }
    }

    // ---- Epilogue. C/D layout: VGPR i -> row wrow+i (lanes 0-15) or +8 (lanes 16-31)
#pragma unroll
    for (int c = 0; c < 4; ++c) {
        int col  = n0 + c * 16 + l16;
        float bv = bias[col];
#pragma unroll
        for (int i = 0; i < 8; ++i) {
            int row = m0 + wrow + i + 8 * half;
            if (row < M) {
                size_t o = (size_t)row * D + col;
                float v  = acc[c][i] + bv;
                if (RELU) v = fmaxf(v, 0.0f);
                if (ACC)  v += C[o];
                C[o] = v;
            }
        }
    }
}

// ---------------------------------------------------------------------------
extern "C" void kernel_launch(void* const* d_in, const int* in_sizes, int n_in,
                              void* d_out, int out_size, void* d_ws, size_t ws_size,
                              hipStream_t stream) {
    const float* x      = (const float*)d_in[0];
    const int*   eidx   = (const int*)d_in[1];   // [2, E]
    const int*   etype  = (const int*)d_in[2];   // [E]
    const float* W_self = (const float*)d_in[3];
    const float* b_self = (const float*)d_in[4];
    const float* W1     = (const float*)d_in[5]; // [3, D, D]
    const float* b1     = (const float*)d_in[6]; // [3, D]
    const float* W2     = (const float*)d_in[7];
    const float* b2     = (const float*)d_in[8];
    float*       out    = (float*)d_out;

    float* agg = (float*)d_ws;                   // N*D f32
    float* h1  = agg + (size_t)N_NODES * D;      // N*D f32

    const int* src = eidx;
    const int* dst = eidx + N_EDGES;

    dim3 ggrid(D / 64, (N_NODES + 127) / 128);   // 8 x 391
    const int n4 = N_NODES * D / 4;

    // out = x @ W_self + b_self
    gemm_kernel<false, false><<<ggrid, 256, 0, stream>>>(x, W_self, b_self, out, N_NODES);

    for (int r = 0; r < NREL; ++r) {
        // agg = x (+) segment_sum of x[src] over relation-r edges
        copy_kernel<<<(n4 + 255) / 256, 256, 0, stream>>>(x, agg, n4);
        scatter_kernel<<<N_EDGES, 256, 0, stream>>>(x, src, dst, etype, r, agg);
        // h1 = relu(agg @ W1_r + b1_r)
        gemm_kernel<true, false><<<ggrid, 256, 0, stream>>>(
            agg, W1 + (size_t)r * D * D, b1 + (size_t)r * D, h1, N_NODES);
        // out += h1 @ W2_r + b2_r
        gemm_kernel<false, true><<<ggrid, 256, 0, stream>>>(
            h1, W2 + (size_t)r * D * D, b2 + (size_t)r * D, out, N_NODES);
    }
}